// decoder_69243462746558
// MI455X (gfx1250) — compile-verified
//
#include <hip/hip_runtime.h>

#define FEAT   512
#define NCLS   64
#define HID    256
#define G4     1024   // 4*HID
#define BATCH  256
#define SEQ    512
#define BT     16     // batch rows per workgroup
#define INDIM  576    // FEAT + NCLS
#define KTOT   768    // FEAT + HID (packed GEMM K)

#define PANEL_K  16
#define PANEL_LD 18                    // 16 + 2 pad floats (bank-conflict-free)
#define PANEL_F  (G4 * PANEL_LD)       // 18432 floats = 72 KB
#define NPANEL   (KTOT / PANEL_K)      // 48

#define XLD   516                      // 512 + TDM pad (2 dwords per 256)
#define HLD   260                      // 256 + 4 pad
#define GLD   1028                     // 1024 + 4 pad

typedef float v2f __attribute__((ext_vector_type(2)));
typedef float v8f __attribute__((ext_vector_type(8)));
typedef unsigned int u32;
typedef unsigned long long u64;
typedef u32 u32x4 __attribute__((ext_vector_type(4)));
typedef u32 u32x8 __attribute__((ext_vector_type(8)));

// ---------------------------------------------------------------------------
// Workspace layout (floats):
//   Wtp  [48][1024][18] : n-major K-panels; Wtp[p][n][kk] = Wt[p*16+kk][n]
//                         where Wt[k][n] = k<512 ? W_ih[n][k] : W_hh[n][k-512]
//   Wo   [64][1024]     : Wo[c][n] = W_ih[n][512+c]   (one-hot columns)
//   bias [1024]         : b_ih + b_hh
//   lossp[16]           : per-workgroup loss partials
// ---------------------------------------------------------------------------
#define WTP_OFF  0
#define WTP_SZ   (NPANEL * PANEL_F)          // 884736
#define WO_OFF   (WTP_OFF + WTP_SZ)
#define BIAS_OFF (WO_OFF + NCLS * G4)        // +65536
#define LOSS_OFF (BIAS_OFF + G4)             // +1024
#define WS_TOTAL (LOSS_OFF + 16)             // 951312 floats (~3.8 MB)

__global__ void pack_kernel(const float* __restrict__ W_ih,
                            const float* __restrict__ W_hh,
                            const float* __restrict__ b_ih,
                            const float* __restrict__ b_hh,
                            float* __restrict__ ws) {
    int idx = blockIdx.x * blockDim.x + threadIdx.x;
    if (idx >= LOSS_OFF) return;
    if (idx < WTP_SZ) {                       // Wtp panels (n-major, padded)
        int p   = idx / PANEL_F;
        int rem = idx - p * PANEL_F;
        int n   = rem / PANEL_LD;
        int kk  = rem - n * PANEL_LD;
        float v = 0.0f;
        if (kk < PANEL_K) {
            int k = p * PANEL_K + kk;
            v = (k < FEAT) ? W_ih[n * INDIM + k] : W_hh[n * HID + (k - FEAT)];
        }
        ws[idx] = v;
    } else if (idx < BIAS_OFF) {              // Wo
        int r = idx - WO_OFF;
        int c = r >> 10, n = r & (G4 - 1);
        ws[idx] = W_ih[n * INDIM + FEAT + c];
    } else {                                  // bias
        int n = idx - BIAS_OFF;
        ws[idx] = b_ih[n] + b_hh[n];
    }
}

__global__ void reduce_kernel(const float* __restrict__ ws, float* __restrict__ out) {
    float s = 0.0f;
    for (int i = 0; i < 16; ++i) s += ws[LOSS_OFF + i];
    out[0] = s;
}

// TDM: contiguous 1D copy of nelem f32 from global to LDS.
__device__ __forceinline__ void tdm_load_1d(const float* gsrc, u32 lds_off, u32 nelem) {
    u64 ga = (u64)gsrc;
    u32x4 g0;
    g0[0] = 1u;
    g0[1] = lds_off;
    g0[2] = (u32)ga;
    g0[3] = ((u32)(ga >> 32) & 0x01FFFFFFu) | (2u << 30);
    u32x8 g1;
    g1[0] = (2u << 16);                              // data_size = 4B
    g1[1] = (nelem & 0xFFFFu) << 16;                 // tensor_dim0 lo16
    g1[2] = ((nelem >> 16) & 0xFFFFu) | (1u << 16);  // tensor_dim0 hi16 | tensor_dim1=1
    g1[3] = (nelem & 0xFFFFu) << 16;                 // tile_dim0 (<=65535)
    g1[4] = 1u;                                      // tile_dim1 = 1
    g1[5] = nelem;                                   // tensor_dim0_stride
    g1[6] = 0u;  g1[7] = 0u;
    u32x4 gz = {0u, 0u, 0u, 0u};
    asm volatile("tensor_load_to_lds %0, %1, %2, %3"
                 :: "s"(g0), "s"(g1), "s"(gz), "s"(gz) : "memory");
}

// TDM: BT x FEAT f32 tile, global row stride SEQ*FEAT, LDS rows padded to 516
// via pad_enable (interval 256 dwords, amount 2 dwords).
__device__ __forceinline__ void tdm_load_x_tile(const float* gsrc, u32 lds_off) {
    u64 ga = (u64)gsrc;
    u32x4 g0;
    g0[0] = 1u;
    g0[1] = lds_off;
    g0[2] = (u32)ga;
    g0[3] = ((u32)(ga >> 32) & 0x01FFFFFFu) | (2u << 30);
    u32x8 g1;
    g1[0] = (2u << 16) | (1u << 20) | (7u << 22) | (1u << 25); // 4B|pad_en|intvl=256dw|amt=2dw
    g1[1] = ((u32)FEAT & 0xFFFFu) << 16;
    g1[2] = (((u32)FEAT >> 16) & 0xFFFFu) | (((u32)BT & 0xFFFFu) << 16);
    g1[3] = (((u32)BT >> 16) & 0xFFFFu) | (((u32)FEAT & 0xFFFFu) << 16);
    g1[4] = (u32)BT;
    g1[5] = (u32)(SEQ * FEAT);
    g1[6] = 0u;  g1[7] = 0u;
    u32x4 gz = {0u, 0u, 0u, 0u};
    asm volatile("tensor_load_to_lds %0, %1, %2, %3"
                 :: "s"(g0), "s"(g1), "s"(gz), "s"(gz) : "memory");
}

__device__ __forceinline__ v8f wmma_f32(v2f a, v2f b, v8f c) {
    return __builtin_amdgcn_wmma_f32_16x16x4_f32(false, a, false, b,
                                                 (short)0, c, false, false);
}

__global__ void lstm_decoder_kernel(const float* __restrict__ x,
                                    const int*   __restrict__ tag,
                                    const float* __restrict__ ws,
                                    const float* __restrict__ W_out,
                                    const float* __restrict__ b_out,
                                    float* __restrict__ wsloss) {
    const float* Wtp  = ws + WTP_OFF;
    const float* Wo   = ws + WO_OFF;
    const float* bias = ws + BIAS_OFF;

    extern __shared__ float smem[];
    float* panelS = smem;                          // [2][PANEL_F]  144 KB
    float* xbufS  = panelS + 2 * PANEL_F;          // [BT][516]     32.25 KB
    float* hbufS  = xbufS + BT * XLD;              // [BT][260]     16.25 KB
    float* gatesS = hbufS + BT * HLD;              // [BT][1028]    64.25 KB
    float* logitS = gatesS + BT * GLD;             // [BT][64]       4 KB
    int*   predS  = (int*)(logitS + BT * NCLS);    // [BT]

    const int tid  = threadIdx.x;
    const int lane = tid & 31;
    const int wv   = tid >> 5;
    const int rowA = lane & 15;                // A-frag matrix row
    const int koff = (lane >> 4) << 1;         // frag K sub-offset: 0 or 2
    const int ncol = lane & 15;                // B/C frag column within tile
    const int mhi  = (lane >> 4) << 3;         // C frag row offset: 0 or 8

    const int b0 = blockIdx.x * BT;

    // ---- init state ----
    for (int i = tid; i < BT * HLD; i += blockDim.x) hbufS[i] = 0.0f;
    if (tid < BT) predS[tid] = 0;
    float creg[16];
#pragma unroll
    for (int i = 0; i < 16; ++i) creg[i] = 0.0f;
    float loss_acc = 0.0f;

    // prologue: x tile 0, then panel pair {0,1}
    if (wv == 0) {
        tdm_load_x_tile(x + (u64)b0 * SEQ * FEAT, (u32)(u64)xbufS);
        tdm_load_1d(Wtp,            (u32)(u64)panelS,              PANEL_F);
        tdm_load_1d(Wtp + PANEL_F,  (u32)(u64)(panelS + PANEL_F),  PANEL_F);
    }

    const float* xa = xbufS + rowA * XLD;
    const float* ha = hbufS + rowA * HLD;

    for (int t = 0; t < SEQ; ++t) {
        v8f acc[8];
#pragma unroll
        for (int j = 0; j < 8; ++j) acc[j] = (v8f)(0.0f);

        // one panel-pair interval: 64 WMMAs entirely from LDS
        auto run_interval = [&](int kb, const float* abase, bool isx) {
#pragma unroll
            for (int half = 0; half < 2; ++half) {
                const float* pan = panelS + half * PANEL_F;
#pragma unroll
                for (int lk0 = 0; lk0 < PANEL_K; lk0 += 4) {
                    int lkk = lk0 + koff;                 // 0..14 within panel
                    int ak  = kb + PANEL_K * half + lkk;  // k within A source
                    int ai  = isx ? (ak + 2 * (ak >> 8)) : ak;  // x-pad skip
                    v2f a = *(const v2f*)&abase[ai];
#pragma unroll
                    for (int j = 0; j < 8; ++j) {
                        int n = (wv * 8 + j) * 16 + ncol;
                        v2f b = *(const v2f*)&pan[n * PANEL_LD + lkk];
                        acc[j] = wmma_f32(a, b, acc[j]);
                    }
                }
            }
        };

        // ---- intervals 0..15: K = 0..511, A from x tile ----
#pragma unroll 1
        for (int i = 0; i < 16; ++i) {
            if (wv == 0) __builtin_amdgcn_s_wait_tensorcnt(0);
            __syncthreads();
            run_interval(32 * i, xa, true);
            __syncthreads();
            if (wv == 0) {   // prefetch next pair (panels 2i+2, 2i+3)
                int p = 2 * i + 2;
                tdm_load_1d(Wtp + (u64)p * PANEL_F, (u32)(u64)panelS, PANEL_F);
                tdm_load_1d(Wtp + (u64)(p + 1) * PANEL_F,
                            (u32)(u64)(panelS + PANEL_F), PANEL_F);
            }
        }
        // ---- intervals 16..23: K = 512..767, A from h ----
#pragma unroll 1
        for (int i = 16; i < 24; ++i) {
            if (wv == 0) __builtin_amdgcn_s_wait_tensorcnt(0);
            __syncthreads();
            run_interval(32 * i - FEAT, ha, false);
            __syncthreads();
            if (wv == 0) {
                if (i < 23) {
                    int p = 2 * i + 2;
                    tdm_load_1d(Wtp + (u64)p * PANEL_F, (u32)(u64)panelS, PANEL_F);
                    tdm_load_1d(Wtp + (u64)(p + 1) * PANEL_F,
                                (u32)(u64)(panelS + PANEL_F), PANEL_F);
                } else if (t + 1 < SEQ) {   // wrap: x tile t+1, panels {0,1}
                    tdm_load_x_tile(x + ((u64)b0 * SEQ + (u64)(t + 1)) * FEAT,
                                    (u32)(u64)xbufS);
                    tdm_load_1d(Wtp,           (u32)(u64)panelS,             PANEL_F);
                    tdm_load_1d(Wtp + PANEL_F, (u32)(u64)(panelS + PANEL_F), PANEL_F);
                }
            }
        }

        // ---- epilogue: bias + one-hot gather + activation -> gates LDS ----
#pragma unroll
        for (int j = 0; j < 8; ++j) {
            int n = (wv * 8 + j) * 16 + ncol;
            int gate = n >> 8;                    // 0:i 1:f 2:g 3:o
            float bs = bias[n];
#pragma unroll
            for (int r = 0; r < 8; ++r) {
                int m = r + mhi;
                float v = acc[j][r] + bs + Wo[predS[m] * G4 + n];
                v = (gate == 2) ? tanhf(v) : (1.0f / (1.0f + expf(-v)));
                gatesS[m * GLD + n] = v;
            }
        }
        __syncthreads();

        // ---- cell update: thread -> (bb = tid&15, j0 = (tid>>4)*16) ----
        {
            int bb = tid & 15;
            int j0 = (tid >> 4) << 4;
#pragma unroll
            for (int i = 0; i < 16; ++i) {
                int j = j0 + i;
                float ig = gatesS[bb * GLD + j];
                float fg = gatesS[bb * GLD + HID + j];
                float gg = gatesS[bb * GLD + 2 * HID + j];
                float og = gatesS[bb * GLD + 3 * HID + j];
                float cn = fg * creg[i] + ig * gg;
                creg[i] = cn;
                hbufS[bb * HLD + j] = og * tanhf(cn);
            }
        }
        __syncthreads();

        // ---- projection: [16,256]x[256,64]; B blocks pre-loaded to regs ----
        if (wv < 4) {
            v8f pacc = (v8f)(0.0f);
            int c0 = wv * 16;
            const float* wob = W_out + (u64)(c0 + ncol) * HID;
#pragma unroll 1
            for (int kb = 0; kb < HID; kb += 32) {
                v2f bb[8];
#pragma unroll
                for (int u = 0; u < 8; ++u)
                    bb[u] = *(const v2f*)&wob[kb + 4 * u + koff];
#pragma unroll
                for (int u = 0; u < 8; ++u) {
                    int kk = kb + 4 * u + koff;
                    v2f a = *(const v2f*)&hbufS[rowA * HLD + kk];
                    pacc = wmma_f32(a, bb[u], pacc);
                }
            }
            float bo = b_out[c0 + ncol];
#pragma unroll
            for (int r = 0; r < 8; ++r)
                logitS[(r + mhi) * NCLS + c0 + ncol] = pacc[r] + bo;
        }
        __syncthreads();

        // ---- argmax + log-softmax + CE (one lane per batch row) ----
        if (tid < BT) {
            int bb = tid;
            float mx = -3.0e38f; int am = 0;
            for (int c = 0; c < NCLS; ++c) {
                float l = logitS[bb * NCLS + c];
                if (l > mx) { mx = l; am = c; }
            }
            float se = 0.0f;
            for (int c = 0; c < NCLS; ++c)
                se += expf(logitS[bb * NCLS + c] - mx);
            float lse = mx + logf(se);
            int ti = tag[(b0 + bb) * SEQ + t];
            if (ti != -1) loss_acc += lse - logitS[bb * NCLS + ti];
            predS[bb] = am;
        }
        __syncthreads();
    }

    // ---- deterministic loss reduction: per-WG partial to workspace ----
    if (tid < BT) logitS[tid] = loss_acc;
    __syncthreads();
    if (tid == 0) {
        float s = 0.0f;
        for (int i = 0; i < BT; ++i) s += logitS[i];
        wsloss[blockIdx.x] = s;
    }
}

extern "C" void kernel_launch(void* const* d_in, const int* in_sizes, int n_in,
                              void* d_out, int out_size, void* d_ws, size_t ws_size,
                              hipStream_t stream) {
    const float* x     = (const float*)d_in[0];
    const int*   tag   = (const int*)d_in[1];
    const float* W_ih  = (const float*)d_in[2];
    const float* W_hh  = (const float*)d_in[3];
    const float* b_ih  = (const float*)d_in[4];
    const float* b_hh  = (const float*)d_in[5];
    const float* W_out = (const float*)d_in[6];
    const float* b_out = (const float*)d_in[7];
    float* out = (float*)d_out;
    float* ws  = (float*)d_ws;

    pack_kernel<<<(LOSS_OFF + 255) / 256, 256, 0, stream>>>(W_ih, W_hh, b_ih, b_hh, ws);

    size_t lds_bytes = (size_t)(2 * PANEL_F + BT * XLD + BT * HLD + BT * GLD
                                + BT * NCLS) * 4 + BT * sizeof(int);  // 267072 B
    lstm_decoder_kernel<<<BATCH / BT, 256, lds_bytes, stream>>>(
        x, tag, ws, W_out, b_out, ws + LOSS_OFF);

    reduce_kernel<<<1, 1, 0, stream>>>(ws, out);
}